// CustomCIFAR10Model_88098369175760
// MI455X (gfx1250) — compile-verified
//
#include <hip/hip_runtime.h>

typedef float v2f __attribute__((ext_vector_type(2)));
typedef float v8f __attribute__((ext_vector_type(8)));

#define D_DIM 3072
#define B_SZ  512
#define NCLS  100
#define ROWBLK 64          // row blocks in phase-1 reduction
#define ROWS_PER_BLK 48    // 64 * 48 = 3072 rows

// ---------------------------------------------------------------------------
// Phase 1: partial column sums of a and b. Grid (3, 64), block 256.
// Each thread owns 4 consecutive columns (float4, fully coalesced 4KB/row/block)
// and reduces 48 rows. 192 blocks * 256 threads keeps enough loads in flight
// to approach the 23.3 TB/s HBM roofline (this is ~91% of all bytes moved).
// ---------------------------------------------------------------------------
__global__ __launch_bounds__(256) void colsum_partial(
    const float* __restrict__ a, const float* __restrict__ b,
    float* __restrict__ pa, float* __restrict__ pb)
{
    const int d4 = blockIdx.x * 1024 + threadIdx.x * 4;   // column group
    const int r0 = blockIdx.y * ROWS_PER_BLK;             // first row
    const float* ap = a + (size_t)r0 * D_DIM + d4;
    const float* bp = b + (size_t)r0 * D_DIM + d4;
    float4 sa = make_float4(0.f, 0.f, 0.f, 0.f);
    float4 sb = make_float4(0.f, 0.f, 0.f, 0.f);
#pragma unroll 8
    for (int i = 0; i < ROWS_PER_BLK; ++i) {
        float4 va = *(const float4*)(ap + (size_t)i * D_DIM);
        float4 vb = *(const float4*)(bp + (size_t)i * D_DIM);
        sa.x += va.x; sa.y += va.y; sa.z += va.z; sa.w += va.w;
        sb.x += vb.x; sb.y += vb.y; sb.z += vb.z; sb.w += vb.w;
    }
    *(float4*)(pa + (size_t)blockIdx.y * D_DIM + d4) = sa;
    *(float4*)(pb + (size_t)blockIdx.y * D_DIM + d4) = sb;
}

// ---------------------------------------------------------------------------
// Phase 2: fold the 64 partials per column. 12 blocks * 256 threads. Trivial.
// Deterministic (fixed summation order) unlike a float-atomic approach.
// ---------------------------------------------------------------------------
__global__ __launch_bounds__(256) void colsum_final(
    const float* __restrict__ pa, const float* __restrict__ pb,
    float* __restrict__ As, float* __restrict__ Bs)
{
    const int d = blockIdx.x * 256 + threadIdx.x;
    float sa = 0.f, sb = 0.f;
#pragma unroll 8
    for (int r = 0; r < ROWBLK; ++r) {
        sa += pa[(size_t)r * D_DIM + d];
        sb += pb[(size_t)r * D_DIM + d];
    }
    As[d] = sa;
    Bs[d] = sb;
}

// ---------------------------------------------------------------------------
// Fused KAN-feature + classifier GEMM with fp32 WMMA (V_WMMA_F32_16X16X4_F32).
//   out[m,n] = sum_d Z[m,d] * fc_w[n,d] + fc_b[n]
//   Z[m,d]   = part1 + cos(x[m,d])*Asum[d] + sin(x[m,d])*Bsum[d]  (built on the fly)
// One wave32 per 16x16 output tile. Grid (512/16, ceil(100/16)) = (32, 7).
// A operand layout (16x4 f32): lane = M (lo 4 bits), VGPR0/1 = K, K+1;
// upper half-wave covers K+2, K+3. Two accumulators (k and k+4) break the
// WMMA D->C dependency chain; summed at the end.
// ---------------------------------------------------------------------------
__global__ __launch_bounds__(32) void kan_fc_wmma(
    const float* __restrict__ xf, const float* __restrict__ As,
    const float* __restrict__ Bs, const float* __restrict__ w,
    const float* __restrict__ npar, const float* __restrict__ fc_w,
    const float* __restrict__ fc_b, float* __restrict__ out)
{
    const int lane  = threadIdx.x;          // 0..31 (wave32)
    const int half  = lane >> 4;            // 0: K/K+1, 1: K+2/K+3
    const int lo    = lane & 15;            // M index (A) / N index (B,C)
    const int mtile = blockIdx.x;
    const int ntile = blockIdx.y;

    const int brow  = mtile * 16 + lo;          // batch row, always < 512
    const int ncol  = ntile * 16 + lo;          // class column, may be >= 100
    const float nmask = (ncol < NCLS) ? 1.0f : 0.0f;  // keep EXEC uniform
    const int   nsafe = (ncol < NCLS) ? ncol : (NCLS - 1);

    // part1: scalar term from the 5-element w / n vectors
    float part1 = 0.f;
#pragma unroll
    for (int i = 0; i < 4; ++i)
        part1 += w[i + 1] * npar[i + 1] + w[i] * npar[i];

    const float* xrow = xf   + (size_t)brow  * D_DIM;
    const float* wrow = fc_w + (size_t)nsafe * D_DIM;

    v8f acc0 = {};
    v8f acc1 = {};

    for (int k = 0; k < D_DIM; k += 8) {
        const int d0 = k + 2 * half;       // this lane's K pair (slot 0)
        const int d1 = d0 + 4;             // this lane's K pair (slot 1)

        // ---- slot 0: build A (Z values) and B (fc_w values) ----
        float2 xv = *(const float2*)(xrow + d0);
        float2 av = *(const float2*)(As + d0);
        float2 bv = *(const float2*)(Bs + d0);
        float s0, cc0, s1, cc1;
        __sincosf(xv.x, &s0, &cc0);
        __sincosf(xv.y, &s1, &cc1);
        v2f amat0;
        amat0.x = part1 + cc0 * av.x + s0 * bv.x;
        amat0.y = part1 + cc1 * av.y + s1 * bv.y;
        float2 wv = *(const float2*)(wrow + d0);
        v2f bmat0;
        bmat0.x = wv.x * nmask;
        bmat0.y = wv.y * nmask;
        acc0 = __builtin_amdgcn_wmma_f32_16x16x4_f32(
            false, amat0, false, bmat0, (short)0, acc0, false, false);

        // ---- slot 1 ----
        xv = *(const float2*)(xrow + d1);
        av = *(const float2*)(As + d1);
        bv = *(const float2*)(Bs + d1);
        __sincosf(xv.x, &s0, &cc0);
        __sincosf(xv.y, &s1, &cc1);
        v2f amat1;
        amat1.x = part1 + cc0 * av.x + s0 * bv.x;
        amat1.y = part1 + cc1 * av.y + s1 * bv.y;
        wv = *(const float2*)(wrow + d1);
        v2f bmat1;
        bmat1.x = wv.x * nmask;
        bmat1.y = wv.y * nmask;
        acc1 = __builtin_amdgcn_wmma_f32_16x16x4_f32(
            false, amat1, false, bmat1, (short)0, acc1, false, false);
    }

    // C/D layout: VGPR r holds M = r (lanes 0-15) / M = r + 8 (lanes 16-31),
    // N = lo. Guarded store for the ragged class dimension.
    if (ncol < NCLS) {
        const float bias = fc_b[ncol];
#pragma unroll
        for (int r = 0; r < 8; ++r) {
            const int m = mtile * 16 + 8 * half + r;
            out[(size_t)m * NCLS + ncol] = acc0[r] + acc1[r] + bias;
        }
    }
}

// ---------------------------------------------------------------------------
extern "C" void kernel_launch(void* const* d_in, const int* in_sizes, int n_in,
                              void* d_out, int out_size, void* d_ws, size_t ws_size,
                              hipStream_t stream) {
    const float* x       = (const float*)d_in[0];   // [512,3,32,32] == [512,3072]
    const float* a       = (const float*)d_in[1];   // [3072,3072,1]
    const float* b       = (const float*)d_in[2];   // [3072,3072,1]
    const float* w       = (const float*)d_in[3];   // [5]
    const float* n_param = (const float*)d_in[4];   // [5]
    const float* fc_w    = (const float*)d_in[5];   // [100,3072]
    const float* fc_b    = (const float*)d_in[6];   // [100]
    float*       out     = (float*)d_out;           // [512,100]

    float* ws = (float*)d_ws;                       // ~1.6 MB used
    float* pa = ws;                                 // [64,3072]
    float* pb = ws + (size_t)ROWBLK * D_DIM;        // [64,3072]
    float* As = ws + 2 * (size_t)ROWBLK * D_DIM;    // [3072]
    float* Bs = As + D_DIM;                         // [3072]

    colsum_partial<<<dim3(3, ROWBLK), 256, 0, stream>>>(a, b, pa, pb);
    colsum_final<<<dim3(D_DIM / 256), 256, 0, stream>>>(pa, pb, As, Bs);
    kan_fc_wmma<<<dim3(B_SZ / 16, (NCLS + 15) / 16), 32, 0, stream>>>(
        x, As, Bs, w, n_param, fc_w, fc_b, out);
}